// Model_61735859913241
// MI455X (gfx1250) — compile-verified
//
#include <hip/hip_runtime.h>
#include <math.h>

#define B_    1024
#define M_    32
#define N_    64
#define D_    32
#define NOBJ_ 20
#define NV_   6890
#define K_    (M_ * N_ * 3)   // 6144
#define KR_   (N_ * 3)        // 192 (M-reduced K)
#define WPB_  8               // waves (batches) per block

typedef __attribute__((ext_vector_type(2))) float v2f;
typedef __attribute__((ext_vector_type(8))) float v8f;

// ---------------------------------------------------------------------------
// Kernel 1: reduce pca_components / pca_means over the M (=32) axis.
// compsR[obj][d][t] = sum_m comps[obj][d][m*192 + t]   (t = n*3 + i)
// meansR[obj][t]    = sum_m means[obj][m*192 + t]
// ---------------------------------------------------------------------------
__global__ void reduce_comps_kernel(const float* __restrict__ comps,
                                    const float* __restrict__ means,
                                    float* __restrict__ compsR,
                                    float* __restrict__ meansR) {
  const int nCompsR = NOBJ_ * D_ * KR_;   // 122880
  const int nMeansR = NOBJ_ * KR_;        // 3840
  int tid = blockIdx.x * blockDim.x + threadIdx.x;
  if (tid < nCompsR) {
    int t   = tid % KR_;
    int rem = tid / KR_;
    int d   = rem % D_;
    int obj = rem / D_;
    const float* src = comps + (size_t)(obj * D_ + d) * K_ + t;
    float s = 0.f;
    #pragma unroll
    for (int m = 0; m < M_; ++m) s += src[m * KR_];
    compsR[tid] = s;
  } else if (tid < nCompsR + nMeansR) {
    int id  = tid - nCompsR;
    int t   = id % KR_;
    int obj = id / KR_;
    const float* src = means + (size_t)obj * K_ + t;
    float s = 0.f;
    #pragma unroll
    for (int m = 0; m < M_; ++m) s += src[m * KR_];
    meansR[id] = s;
  }
}

// ---------------------------------------------------------------------------
// wave32 helpers
// ---------------------------------------------------------------------------
__device__ __forceinline__ float wave_sum32(float v) {
  #pragma unroll
  for (int o = 16; o > 0; o >>= 1) v += __shfl_xor(v, o, 32);
  return v;
}

// ---------------------------------------------------------------------------
// Kernel 2: one wave32 per batch, 8 batches per block.
// S = compsR[obj]^T gamma + meansR ; H = S^T Qc via v_wmma_f32_16x16x4_f32 ;
// 3x3 SVD (Kabsch) on lane 0 -> R, T.
// ---------------------------------------------------------------------------
__global__ __launch_bounds__(32 * WPB_)
void per_batch_kernel(const float* __restrict__ gamma,
                      const float* __restrict__ smpl_v,
                      const float* __restrict__ object_anchors,
                      const int*  __restrict__ labels,
                      const int*  __restrict__ aidx,
                      const float* __restrict__ compsR,
                      const float* __restrict__ meansR,
                      float* __restrict__ out) {
  const int wid  = threadIdx.x >> 5;
  const int lane = threadIdx.x & 31;   // full wave active (EXEC all-1s)
  const int b    = blockIdx.x * WPB_ + wid;
  const int obj  = labels[b];

  // Wave-private LDS slices. S_T / Qc_T are transposed (component-major,
  // n contiguous) and carry a zeroed 4th row so WMMA rows 3..15 read zeros
  // with unconditional clamped loads (no EXEC predication).
  __shared__ float g_s [WPB_][D_];
  __shared__ float S_T [WPB_][4][N_];
  __shared__ float Qc_T[WPB_][4][N_];
  __shared__ float H_s [WPB_][9];

  g_s[wid][lane] = gamma[b * D_ + lane];

  // zero the padding rows (row 3) of both transposed arrays
  S_T [wid][3][lane]      = 0.f;
  S_T [wid][3][lane + 32] = 0.f;
  Qc_T[wid][3][lane]      = 0.f;
  Qc_T[wid][3][lane + 32] = 0.f;

  // --- object anchors: centroid + centered cloud (N=64, two rows/lane) ---
  const float* oaBase = object_anchors + (size_t)obj * (N_ * 3);
  float oa0[3], oa1[3], oabar[3];
  #pragma unroll
  for (int i = 0; i < 3; ++i) {
    oa0[i] = oaBase[lane * 3 + i];
    oa1[i] = oaBase[(lane + 32) * 3 + i];
  }
  #pragma unroll
  for (int i = 0; i < 3; ++i)
    oabar[i] = wave_sum32(oa0[i] + oa1[i]) * (1.f / (float)N_);
  #pragma unroll
  for (int i = 0; i < 3; ++i) {
    Qc_T[wid][i][lane]      = oa0[i] - oabar[i];
    Qc_T[wid][i][lane + 32] = oa1[i] - oabar[i];
  }

  // --- SMPL anchor centroid (M=32, one anchor per lane) ---
  const int vidx = aidx[lane];
  float abar[3];
  #pragma unroll
  for (int i = 0; i < 3; ++i) {
    float av = smpl_v[((size_t)b * NV_ + vidx) * 3 + i];
    abar[i] = wave_sum32(av) * (1.f / (float)M_);
  }

  __syncthreads();

  // --- S[t] = sum_d compsR[obj][d][t]*gamma[d] + meansR[obj][t]; 6 t/lane ---
  float s[6];
  {
    const float* mR = meansR + obj * KR_ + lane * 6;
    #pragma unroll
    for (int j = 0; j < 6; ++j) s[j] = mR[j];
    const float* cR = compsR + (size_t)obj * (D_ * KR_) + lane * 6;
    #pragma unroll 4
    for (int d = 0; d < D_; ++d) {
      const float gd = g_s[wid][d];
      const float* row = cR + d * KR_;
      #pragma unroll
      for (int j = 0; j < 6; ++j) s[j] = fmaf(gd, row[j], s[j]);
    }
  }
  // lane covers n = 2*lane (+0) and n = 2*lane+1 (+3): S_T[i][n]
  #pragma unroll
  for (int i = 0; i < 3; ++i) {
    S_T[wid][i][2 * lane]     = s[i];
    S_T[wid][i][2 * lane + 1] = s[i + 3];
  }

  // --- mean offset over all k = (1/2048) * sum_n S[n,:] ---
  float offbar[3];
  #pragma unroll
  for (int i = 0; i < 3; ++i)
    offbar[i] = wave_sum32(s[i] + s[i + 3]) * (1.f / (float)(M_ * N_));

  __syncthreads();

  // --- H(3x3) = S^T(3x64) * Qc(64x3) via WMMA f32 16x16x4, 16 K-chunks ---
  // A 16x4 layout: lanes 0-15 -> row M=lane, v0:K=0 v1:K=1;
  //                lanes 16-31 -> row M=lane-16, v0:K=2 v1:K=3. (ISA 7.12.2)
  // Rows >= 3 clamp to the zeroed pad row -> fragments are exact zeros.
  v8f c = {0.f, 0.f, 0.f, 0.f, 0.f, 0.f, 0.f, 0.f};
  const int row16 = lane & 15;
  const int khalf = lane >> 4;
  const int rowc  = (row16 < 3) ? row16 : 3;
  const float* Sbase = &S_T [wid][rowc][0];
  const float* Qbase = &Qc_T[wid][rowc][0];
  #pragma unroll
  for (int ch = 0; ch < 16; ++ch) {
    const int nb = ch * 4 + 2 * khalf;           // even -> 8B aligned
    const v2f a  = *(const v2f*)(Sbase + nb);
    const v2f bb = *(const v2f*)(Qbase + nb);
    c = __builtin_amdgcn_wmma_f32_16x16x4_f32(false, a, false, bb,
                                              (short)0, c, false, false);
  }
  // C layout: VGPR r holds row M=r (lanes 0-15); H[i][j] = vgpr i @ lane j.
  if (lane < 3) {
    H_s[wid][0 * 3 + lane] = c[0];
    H_s[wid][1 * 3 + lane] = c[1];
    H_s[wid][2 * 3 + lane] = c[2];
  }
  __syncthreads();

  // --- 3x3 SVD / Kabsch on lane 0 of each wave ---
  if (lane == 0) {
    float Hm[3][3];
    #pragma unroll
    for (int i = 0; i < 3; ++i)
      #pragma unroll
      for (int j = 0; j < 3; ++j) Hm[i][j] = H_s[wid][i * 3 + j];

    // A = H^T H (symmetric PSD)
    float A[3][3];
    #pragma unroll
    for (int i = 0; i < 3; ++i)
      #pragma unroll
      for (int j = 0; j < 3; ++j)
        A[i][j] = Hm[0][i] * Hm[0][j] + Hm[1][i] * Hm[1][j] + Hm[2][i] * Hm[2][j];

    float V[3][3] = {{1.f, 0.f, 0.f}, {0.f, 1.f, 0.f}, {0.f, 0.f, 1.f}};

    // Cyclic Jacobi eigensolver, branchless rotations
    for (int sweep = 0; sweep < 12; ++sweep) {
      #pragma unroll
      for (int r = 0; r < 3; ++r) {
        const int p = (r == 2) ? 1 : 0;
        const int q = (r == 0) ? 1 : 2;
        const float apq   = A[p][q];
        const bool  live  = fabsf(apq) > 1e-20f;
        const float apq_s = live ? apq : 1.f;
        const float theta = (A[q][q] - A[p][p]) / (2.f * apq_s);
        float t = copysignf(1.f, theta) /
                  (fabsf(theta) + sqrtf(theta * theta + 1.f));
        t = live ? t : 0.f;                       // dead -> identity rotation
        const float cc = 1.f / sqrtf(t * t + 1.f);
        const float ss = t * cc;
        #pragma unroll
        for (int k = 0; k < 3; ++k) {             // column rotation A*G
          const float akp = A[k][p], akq = A[k][q];
          A[k][p] = cc * akp - ss * akq;
          A[k][q] = ss * akp + cc * akq;
        }
        #pragma unroll
        for (int k = 0; k < 3; ++k) {             // row rotation G^T*A
          const float apk = A[p][k], aqk = A[q][k];
          A[p][k] = cc * apk - ss * aqk;
          A[q][k] = ss * apk + cc * aqk;
        }
        #pragma unroll
        for (int k = 0; k < 3; ++k) {             // V = V*G
          const float vkp = V[k][p], vkq = V[k][q];
          V[k][p] = cc * vkp - ss * vkq;
          V[k][q] = ss * vkp + cc * vkq;
        }
      }
    }

    // sort eigenpairs descending
    float lam[3] = {A[0][0], A[1][1], A[2][2]};
    #pragma unroll
    for (int i = 0; i < 2; ++i)
      #pragma unroll
      for (int j = 0; j < 2; ++j)
        if (lam[j] < lam[j + 1]) {
          float tl = lam[j]; lam[j] = lam[j + 1]; lam[j + 1] = tl;
          #pragma unroll
          for (int k = 0; k < 3; ++k) {
            float tv = V[k][j]; V[k][j] = V[k][j + 1]; V[k][j + 1] = tv;
          }
        }

    // U columns: u0 = norm(H v0); u1 = norm(H v1 orth u0); u2 = u0 x u1.
    // Flipping u2 also flips det -> R invariant, so cross() is sign-safe.
    float U[3][3];
    {
      float x = Hm[0][0] * V[0][0] + Hm[0][1] * V[1][0] + Hm[0][2] * V[2][0];
      float y = Hm[1][0] * V[0][0] + Hm[1][1] * V[1][0] + Hm[1][2] * V[2][0];
      float z = Hm[2][0] * V[0][0] + Hm[2][1] * V[1][0] + Hm[2][2] * V[2][0];
      float inv = 1.f / sqrtf(fmaxf(x * x + y * y + z * z, 1e-30f));
      U[0][0] = x * inv; U[1][0] = y * inv; U[2][0] = z * inv;
    }
    {
      float x = Hm[0][0] * V[0][1] + Hm[0][1] * V[1][1] + Hm[0][2] * V[2][1];
      float y = Hm[1][0] * V[0][1] + Hm[1][1] * V[1][1] + Hm[1][2] * V[2][1];
      float z = Hm[2][0] * V[0][1] + Hm[2][1] * V[1][1] + Hm[2][2] * V[2][1];
      float dp = x * U[0][0] + y * U[1][0] + z * U[2][0];
      x -= dp * U[0][0]; y -= dp * U[1][0]; z -= dp * U[2][0];
      float inv = 1.f / sqrtf(fmaxf(x * x + y * y + z * z, 1e-30f));
      U[0][1] = x * inv; U[1][1] = y * inv; U[2][1] = z * inv;
    }
    U[0][2] = U[1][0] * U[2][1] - U[2][0] * U[1][1];
    U[1][2] = U[2][0] * U[0][1] - U[0][0] * U[2][1];
    U[2][2] = U[0][0] * U[1][1] - U[1][0] * U[0][1];

    // d = det(U V^T) = det(U) det(V); det(U) = +1 by construction
    const float detV =
        V[0][0] * (V[1][1] * V[2][2] - V[1][2] * V[2][1]) -
        V[0][1] * (V[1][0] * V[2][2] - V[1][2] * V[2][0]) +
        V[0][2] * (V[1][0] * V[2][1] - V[1][1] * V[2][0]);
    const float dsgn = (detV >= 0.f) ? 1.f : -1.f;

    // R = U diag(1,1,d) V^T ;  T = pbar - R * q0bar
    float R[3][3];
    #pragma unroll
    for (int i = 0; i < 3; ++i)
      #pragma unroll
      for (int j = 0; j < 3; ++j)
        R[i][j] = U[i][0] * V[j][0] + U[i][1] * V[j][1] + dsgn * U[i][2] * V[j][2];

    float* Rout = out + (size_t)b * 9;
    float* Tout = out + (size_t)B_ * 9 + (size_t)b * 3;
    #pragma unroll
    for (int i = 0; i < 3; ++i) {
      #pragma unroll
      for (int j = 0; j < 3; ++j) Rout[i * 3 + j] = R[i][j];
      const float pb = abar[i] + offbar[i];
      Tout[i] = pb - (R[i][0] * oabar[0] + R[i][1] * oabar[1] + R[i][2] * oabar[2]);
    }
  }
}

// ---------------------------------------------------------------------------
extern "C" void kernel_launch(void* const* d_in, const int* in_sizes, int n_in,
                              void* d_out, int out_size, void* d_ws, size_t ws_size,
                              hipStream_t stream) {
  (void)in_sizes; (void)n_in; (void)out_size; (void)ws_size;
  const float* gamma  = (const float*)d_in[0];
  const float* smpl_v = (const float*)d_in[1];
  const float* comps  = (const float*)d_in[2];
  const float* means  = (const float*)d_in[3];
  const float* oa     = (const float*)d_in[4];
  const int*   labels = (const int*)d_in[5];
  const int*   aidx   = (const int*)d_in[6];
  float* out = (float*)d_out;

  float* compsR = (float*)d_ws;                       // 122880 floats
  float* meansR = compsR + NOBJ_ * D_ * KR_;          //   3840 floats

  const int n1 = NOBJ_ * D_ * KR_ + NOBJ_ * KR_;      // 126720
  reduce_comps_kernel<<<(n1 + 255) / 256, 256, 0, stream>>>(comps, means,
                                                            compsR, meansR);
  per_batch_kernel<<<B_ / WPB_, 32 * WPB_, 0, stream>>>(gamma, smpl_v, oa,
                                                        labels, aidx,
                                                        compsR, meansR, out);
}